// GCNAttention_12919261626425
// MI455X (gfx1250) — compile-verified
//
#include <hip/hip_runtime.h>
#include <hip/hip_bf16.h>

// ---------------------------------------------------------------------------
// Problem constants (match reference)
// ---------------------------------------------------------------------------
#define BATCH   64
#define CAT     100
#define LOOPN   3
#define NBPER   150
#define NTOT    (BATCH * NBPER)   // 9600
#define FEAT    2052
#define OUTF    2048

typedef float v2f __attribute__((ext_vector_type(2)));
typedef float v8f __attribute__((ext_vector_type(8)));

// ---------------------------------------------------------------------------
// Kernel 1: build x[B*C, 2052] = lin_b + sum_l lin_w[slot] * full_features
// ---------------------------------------------------------------------------
__global__ __launch_bounds__(256)
void k_build_x(const float* __restrict__ imgf,   // [N,2048]
               const float* __restrict__ bbox,   // [N,4]
               const int*   __restrict__ label,  // [N]
               const int*   __restrict__ batch,  // [N]
               const float* __restrict__ lin_w,  // [3]
               const float* __restrict__ lin_b,  // [1]
               float*       __restrict__ xbuf)   // [B*C, 2052]
{
    const int blk = blockIdx.x;            // 0 .. B*C-1
    const int b   = blk / CAT;
    const int c   = blk % CAT;

    __shared__ int   midx[LOOPN];
    __shared__ float mw[LOOPN];
    __shared__ int   mcount;

    if (threadIdx.x == 0) {
        int cnt = 0;
        for (int j = 0; j < NTOT; ++j) {
            if (batch[j] == b && (label[j] - 1) == c) {
                if (cnt < LOOPN) { midx[cnt] = j; mw[cnt] = lin_w[cnt]; }
                ++cnt;
            }
        }
        mcount = cnt < LOOPN ? cnt : LOOPN;
    }
    __syncthreads();

    const float bias = lin_b[0];
    const int   nm   = mcount;
    for (int d = threadIdx.x; d < FEAT; d += blockDim.x) {
        float acc = bias;
        for (int m = 0; m < nm; ++m) {
            const int j = midx[m];
            const float v = (d < 2048) ? imgf[(size_t)j * 2048 + d]
                                       : bbox[(size_t)j * 4 + (d - 2048)];
            acc += mw[m] * v;
        }
        xbuf[(size_t)blk * FEAT + d] = acc;
    }
}

// ---------------------------------------------------------------------------
// Kernel 2: support = x @ gc_w     [6400,2052] x [2052,2048] -> [6400,2048]
// 64x64 tile / block, 8 waves, each wave 16 rows x 32 cols (two v8f accums).
// Main loop: 64 guard-free K-chunks of 32, double-buffered LDS staged with
// GLOBAL_LOAD_ASYNC_TO_LDS (ASYNCcnt pipelining). Tail (k=2048..2051) is one
// direct-from-global WMMA step (all accesses in-bounds).
// ---------------------------------------------------------------------------
#define TM   64
#define TN   64
#define KC   32
#define LDA  36                    // row stride (floats): 16B-aligned rows
#define ABUF (TM * LDA)            // 2304 floats
#define BUFN (2 * TM * LDA)        // 4608 floats per double-buffer slot
#define NCH  64                    // full K chunks (64*32 = 2048)

__global__ __launch_bounds__(256)
void k_gemm_support(const float* __restrict__ xbuf,  // [6400,2052]
                    const float* __restrict__ gcw,   // [2052,2048]
                    float*       __restrict__ sup)   // [6400,2048]
{
    __shared__ float smem[2][BUFN];   // [buf][ A(64x36) | B-transposed(64x36) ]

    const int tid   = threadIdx.x;
    const int mBase = blockIdx.y * TM;
    const int nBase = blockIdx.x * TN;
    const int wave  = tid >> 5;
    const int lane  = tid & 31;
    const int lo    = lane & 15;
    const int hi    = lane >> 4;
    const int mOff  = (wave >> 1) * 16;   // 0,16,32,48
    const int nOff  = (wave & 1) * 32;    // 0,32

    // ---- per-thread staging offsets (constant across chunks) ----
    // A: 2 x b128 per thread: element e = tid + i*256 -> row e/8, k4 = (e%8)*4
    unsigned aVoff[2], aLds[2];
    // B: 8 x b32 per thread: element e = tid + i*256 -> n = e%64, kk = e/64
    unsigned bVoff[8], bLds[8];
    const unsigned smemBase = (unsigned)(unsigned long long)&smem[0][0];
    #pragma unroll
    for (int i = 0; i < 2; ++i) {
        const int e  = tid + i * 256;
        const int r  = e >> 3;
        const int k4 = (e & 7) * 4;
        aVoff[i] = (unsigned)(((mBase + r) * FEAT + k4) * 4);
        aLds[i]  = (unsigned)((r * LDA + k4) * 4);
    }
    #pragma unroll
    for (int i = 0; i < 8; ++i) {
        const int e  = tid + i * 256;
        const int n  = e & 63;
        const int kk = e >> 6;
        bVoff[i] = (unsigned)((kk * OUTF + nBase + n) * 4);
        bLds[i]  = (unsigned)((ABUF + n * LDA + kk) * 4);
    }

    v8f acc0 = {0.f,0.f,0.f,0.f,0.f,0.f,0.f,0.f};
    v8f acc1 = {0.f,0.f,0.f,0.f,0.f,0.f,0.f,0.f};

    // issue one chunk's async loads into buffer `buf`
    auto issue_chunk = [&](int k0, int buf) {
        const char* aS = (const char*)xbuf + (size_t)k0 * 4;          // +k0 floats
        const char* bS = (const char*)gcw  + (size_t)k0 * (OUTF * 4); // +k0 rows
        const unsigned ldsB = smemBase + (unsigned)(buf * (BUFN * 4));
        #pragma unroll
        for (int i = 0; i < 2; ++i)
            asm volatile("global_load_async_to_lds_b128 %0, %1, %2"
                         :: "v"(ldsB + aLds[i]), "v"(aVoff[i]), "s"(aS)
                         : "memory");
        #pragma unroll
        for (int i = 0; i < 8; ++i)
            asm volatile("global_load_async_to_lds_b32 %0, %1, %2"
                         :: "v"(ldsB + bLds[i]), "v"(bVoff[i]), "s"(bS)
                         : "memory");
    };

    // compute one staged chunk from buffer `buf`
    auto compute_chunk = [&](int buf) {
        const float* sA = &smem[buf][0];
        const float* sB = &smem[buf][ABUF];
        #pragma unroll
        for (int kk = 0; kk < KC; kk += 4) {
            const v2f a  = *(const v2f*)&sA[(mOff + lo)      * LDA + kk + 2 * hi];
            const v2f b0 = *(const v2f*)&sB[(nOff + lo)      * LDA + kk + 2 * hi];
            const v2f b1 = *(const v2f*)&sB[(nOff + 16 + lo) * LDA + kk + 2 * hi];
            acc0 = __builtin_amdgcn_wmma_f32_16x16x4_f32(
                       false, a, false, b0, (short)0, acc0, false, false);
            acc1 = __builtin_amdgcn_wmma_f32_16x16x4_f32(
                       false, a, false, b1, (short)0, acc1, false, false);
        }
    };

    issue_chunk(0, 0);                               // prologue
    for (int i = 0; i < NCH; ++i) {
        if (i + 1 < NCH) {
            issue_chunk((i + 1) * KC, (i + 1) & 1);  // prefetch next chunk
            // in-order completion: <=10 outstanding => chunk i landed
            asm volatile("s_wait_asynccnt 0xa" ::: "memory");
        } else {
            asm volatile("s_wait_asynccnt 0x0" ::: "memory");
        }
        __syncthreads();          // all waves' chunk-i data visible
        compute_chunk(i & 1);
        __syncthreads();          // safe to overwrite this buffer next round
    }

    // ---- K tail: k = 2048..2051, direct from global (all in-bounds) ----
    {
        const int ka = 2048 + 2 * hi;                 // <= 2050, ka+1 <= 2051
        v2f a, b0, b1;
        const size_t arow = (size_t)(mBase + mOff + lo) * FEAT;
        a.x  = xbuf[arow + ka];
        a.y  = xbuf[arow + ka + 1];
        b0.x = gcw[(size_t)ka * OUTF + nBase + nOff + lo];
        b0.y = gcw[(size_t)(ka + 1) * OUTF + nBase + nOff + lo];
        b1.x = gcw[(size_t)ka * OUTF + nBase + nOff + 16 + lo];
        b1.y = gcw[(size_t)(ka + 1) * OUTF + nBase + nOff + 16 + lo];
        acc0 = __builtin_amdgcn_wmma_f32_16x16x4_f32(
                   false, a, false, b0, (short)0, acc0, false, false);
        acc1 = __builtin_amdgcn_wmma_f32_16x16x4_f32(
                   false, a, false, b1, (short)0, acc1, false, false);
    }

    // D layout: VGPR i -> row (i + 8*hi), col = lane%16
    #pragma unroll
    for (int i = 0; i < 8; ++i) {
        const int m = mBase + mOff + i + 8 * hi;
        sup[(size_t)m * OUTF + nBase + nOff + lo]      = acc0[i];
        sup[(size_t)m * OUTF + nBase + nOff + 16 + lo] = acc1[i];
    }
}

// ---------------------------------------------------------------------------
// Kernel 3: tmp[b,c,o] = leaky_relu( sum_n (X[c,n]+adj[b,c,n]) * sup[b,n,o]
//                                    + gc_b[o] )    (f32 WMMA, L2-resident)
// ---------------------------------------------------------------------------
__global__ __launch_bounds__(256)
void k_gcn(const float* __restrict__ Xp,   // [100,100]
           const float* __restrict__ adj,  // [64,100,100]
           const float* __restrict__ sup,  // [6400,2048]
           const float* __restrict__ gcb,  // [2048]
           float*       __restrict__ tmp)  // [6400,2048]
{
    const int b    = blockIdx.y;
    const int wave = threadIdx.x >> 5;
    const int lane = threadIdx.x & 31;
    const int t    = blockIdx.x * 8 + wave;     // 0..895 (7 m-tiles x 128 n-tiles)
    const int m0   = (t / 128) * 16;
    const int n0   = (t % 128) * 16;
    const int lo   = lane & 15;
    const int hi   = lane >> 4;

    v8f acc = {0.f,0.f,0.f,0.f,0.f,0.f,0.f,0.f};

    const int  m  = m0 + lo;
    const bool mv = (m < CAT);

    for (int k = 0; k < CAT; k += 4) {
        const int ka = k + 2 * hi;
        v2f a, bf;
        a.x = mv ? (Xp[m * CAT + ka]     + adj[((size_t)b * CAT + m) * CAT + ka])     : 0.f;
        a.y = mv ? (Xp[m * CAT + ka + 1] + adj[((size_t)b * CAT + m) * CAT + ka + 1]) : 0.f;
        bf.x = sup[((size_t)b * CAT + ka)     * OUTF + n0 + lo];
        bf.y = sup[((size_t)b * CAT + ka + 1) * OUTF + n0 + lo];
        acc = __builtin_amdgcn_wmma_f32_16x16x4_f32(
                  false, a, false, bf, (short)0, acc, false, false);
    }

    const float bias = gcb[n0 + lo];
    #pragma unroll
    for (int i = 0; i < 8; ++i) {
        const int mm = m0 + i + 8 * hi;
        if (mm < CAT) {
            float v = acc[i] + bias;
            v = (v < 0.f) ? 0.01f * v : v;   // leaky_relu, slope 0.01
            tmp[((size_t)b * CAT + mm) * OUTF + n0 + lo] = v;
        }
    }
}

// ---------------------------------------------------------------------------
// Kernel 4: out[b,o] = sum_c global_features[b,c] * tmp[b,c,o]
// ---------------------------------------------------------------------------
__global__ __launch_bounds__(256)
void k_attend(const float* __restrict__ gfeat,  // [64,100]
              const float* __restrict__ tmp,    // [6400,2048]
              float*       __restrict__ out)    // [64,2048]
{
    const int b = blockIdx.y;
    const int o = blockIdx.x * 256 + threadIdx.x;
    float acc = 0.f;
    for (int c = 0; c < CAT; ++c)
        acc += gfeat[b * CAT + c] * tmp[((size_t)b * CAT + c) * OUTF + o];
    out[(size_t)b * OUTF + o] = acc;
}

// ---------------------------------------------------------------------------
// Host-side launcher
// ---------------------------------------------------------------------------
extern "C" void kernel_launch(void* const* d_in, const int* in_sizes, int n_in,
                              void* d_out, int out_size, void* d_ws, size_t ws_size,
                              hipStream_t stream) {
    const float* imgf  = (const float*)d_in[0];   // [9600,2048]
    const float* bbox  = (const float*)d_in[1];   // [9600,4]
    const float* gfeat = (const float*)d_in[2];   // [64,100]
    const float* adj   = (const float*)d_in[3];   // [64,100,100]
    const float* Xp    = (const float*)d_in[4];   // [100,100]
    const float* lin_w = (const float*)d_in[5];   // [3]
    const float* lin_b = (const float*)d_in[6];   // [1]
    const float* gcw   = (const float*)d_in[7];   // [2052,2048]
    const float* gcb   = (const float*)d_in[8];   // [2048]
    const int*   label = (const int*)d_in[9];     // [9600]
    const int*   batch = (const int*)d_in[10];    // [9600]
    float*       out   = (float*)d_out;           // [64,2048]

    float* xbuf = (float*)d_ws;                         // [6400, 2052]
    float* sup  = xbuf + (size_t)BATCH * CAT * FEAT;    // [6400, 2048]
    float* tmp  = xbuf;  // x fully consumed by k_gemm_support; safe to reuse

    // 1) build x
    k_build_x<<<dim3(BATCH * CAT), dim3(256), 0, stream>>>(
        imgf, bbox, label, batch, lin_w, lin_b, xbuf);

    // 2) support = x @ gc_w   (6400 x 2052 x 2048, f32 WMMA, async-LDS staged)
    k_gemm_support<<<dim3(OUTF / TN, (BATCH * CAT) / TM), dim3(256), 0, stream>>>(
        xbuf, gcw, sup);

    // 3) tmp = leaky_relu((X + adj) @ support + gc_b)
    k_gcn<<<dim3(112, BATCH), dim3(256), 0, stream>>>(
        Xp, adj, sup, gcb, tmp);

    // 4) out = global_features-weighted sum over categories
    k_attend<<<dim3(OUTF / 256, BATCH), dim3(256), 0, stream>>>(
        gfeat, tmp, out);
}